// SSIMLoss_59854664237599
// MI455X (gfx1250) — compile-verified
//
#include <hip/hip_runtime.h>

// SSIM loss for img shape (16, 3, 512, 512) fp32.
// Separable 11x11 Gaussian depthwise conv expressed as banded matmuls on
// V_WMMA_F32_16X16X4_F32. One wave32 per 16x16 output tile.

typedef __attribute__((ext_vector_type(2))) float v2f;
typedef __attribute__((ext_vector_type(4))) float v4f;
typedef __attribute__((ext_vector_type(8))) float v8f;

#define WMMA4(A, B, C) \
    __builtin_amdgcn_wmma_f32_16x16x4_f32(false, (A), false, (B), (short)0, (C), false, false)

#define IMG_H 512
#define IMG_W 512
#define PLANES 48           // 16 images * 3 channels
#define TILES_X 32
#define TILES_Y 32
#define NBLOCKS (PLANES * TILES_Y * TILES_X)   // 49152
#define NPIX 12582912.0     // 16*3*512*512

#define TSTR 28             // halo tile row stride (26 cols + 2 zero pad, even)
#define TROWS 26
#define CSTR 28             // T column stride (112B: 16B aligned, conflict-free)

__global__ void ssim_init_kernel(double* acc) {
    if (threadIdx.x == 0 && blockIdx.x == 0) acc[0] = 0.0;
}

__global__ void ssim_final_kernel(const double* acc, float* out) {
    if (threadIdx.x == 0 && blockIdx.x == 0)
        out[0] = 1.0f - (float)(acc[0] / NPIX);
}

// Clamp-index LDS lookup + validity mask: med3 + ds_load + cndmask, no branches.
__device__ __forceinline__ float gco(const float* gt, int d) {
    int idx = d < 0 ? 0 : (d > 10 ? 10 : d);
    float v = gt[idx];
    return ((unsigned)d < 11u) ? v : 0.0f;
}

__global__ __launch_bounds__(32)
void ssim_main_kernel(const float* __restrict__ img1,
                      const float* __restrict__ img2,
                      double* __restrict__ acc_out) {
    // halo tiles, row-major (stride 28; cols 26,27 zeroed)
    __shared__ __attribute__((aligned(16))) float t1[TROWS * TSTR];
    __shared__ __attribute__((aligned(16))) float t2[TROWS * TSTR];
    // intermediate conv results, COLUMN-major: Tq[q][n*CSTR + k], k=0..25,
    // k=26,27 zero pad so V-pass b64 loads need no guards.
    __shared__ __attribute__((aligned(16))) float Tq[5][16 * CSTR];
    __shared__ __attribute__((aligned(16))) float gtab[12];

    const int l = threadIdx.x;          // 0..31
    const int b = blockIdx.x;
    const int tx = (b & (TILES_X - 1)) * 16;
    const int ty = ((b >> 5) & (TILES_Y - 1)) * 16;
    const int plane = b >> 10;

    const float* p1 = img1 + (size_t)plane * (IMG_H * IMG_W);
    const float* p2 = img2 + (size_t)plane * (IMG_H * IMG_W);

    const int mlo = l & 15;
    const int koff = (l >> 4) << 1;     // 0 for lanes 0-15, 2 for lanes 16-31

    // ---- Normalized Gaussian window (11, sigma=1.5) as literals ----
    // exp(-(i-5)^2/4.5)/sum ; symmetric, sums to 1.0f exactly in fp32.
    {
        const v4f gA = {0.00102838f, 0.00759876f, 0.03600078f, 0.10936069f};
        const v4f gB = {0.21300553f, 0.26601172f, 0.21300553f, 0.10936069f};
        const v4f gC = {0.03600078f, 0.00759876f, 0.00102838f, 0.0f};
        *(v4f*)&gtab[0] = gA;   // all lanes write identical data: benign
        *(v4f*)&gtab[4] = gB;
        *(v4f*)&gtab[8] = gC;
    }

    // ---- Load 26x26 halo tiles (zero padded; pad cols 26,27 zeroed) ----
    for (int idx = l; idx < TROWS * TSTR; idx += 32) {
        int r = idx / TSTR;
        int c = idx - r * TSTR;
        int gy = ty + r - 5;
        int gx = tx + c - 5;
        float v1 = 0.0f, v2 = 0.0f;
        if (c < 26 && gy >= 0 && gy < IMG_H && gx >= 0 && gx < IMG_W) {
            int off = gy * IMG_W + gx;
            v1 = p1[off];
            v2 = p2[off];
        }
        t1[idx] = v1;
        t2[idx] = v2;
    }
    // single wave per block: LDS ops are in-order within the wave, no barrier.

    // wco[c] = banded coefficient pair for K-chunk c; identical expression
    // serves Gh columns (H pass B) and Gv rows (V pass A): g[kb+e - mlo].
    v2f wco[7];
#pragma unroll
    for (int c = 0; c < 7; ++c) {
        int d0 = (c << 2) + koff - mlo;
        wco[c].x = gco(gtab, d0);
        wco[c].y = gco(gtab, d0 + 1);
    }

    const v8f zero8 = {0.f, 0.f, 0.f, 0.f, 0.f, 0.f, 0.f, 0.f};

    // ================= Horizontal pass: T = P * Gh =================
    // out[m][n] = sum_d P[m][n+d] * g[d]   (26 rows -> two 16-row groups)
#pragma unroll
    for (int grp = 0; grp < 2; ++grp) {
        int row = (grp << 4) + mlo;
        if (row > 25) row = 25;                 // clamp; results discarded
        const int rbase = row * TSTR;

        v8f acc[5];
#pragma unroll
        for (int q = 0; q < 5; ++q) acc[q] = zero8;

#pragma unroll
        for (int c = 0; c < 7; ++c) {
            const int kb = (c << 2) + koff;     // even -> 8B aligned LDS loads
            v2f a1 = *(const v2f*)(t1 + rbase + kb);
            v2f a2 = *(const v2f*)(t2 + rbase + kb);
            v2f Aq[5];
            Aq[0] = a1;
            Aq[1] = a2;
            Aq[2] = a1 * a1;
            Aq[3] = a2 * a2;
            Aq[4] = a1 * a2;
#pragma unroll
            for (int q = 0; q < 5; ++q)
                acc[q] = WMMA4(Aq[q], wco[c], acc[q]);
        }

        // C/D layout: VGPR j -> row j (lanes 0-15) / row j+8 (lanes 16-31).
        // Store column-major: consecutive C rows -> consecutive k addresses.
        if (l < 16) {
#pragma unroll
            for (int q = 0; q < 5; ++q) {
                float* col = &Tq[q][mlo * CSTR + (grp << 4)];
                *(v4f*)(col)     = __builtin_shufflevector(acc[q], acc[q], 0, 1, 2, 3);
                *(v4f*)(col + 4) = __builtin_shufflevector(acc[q], acc[q], 4, 5, 6, 7);
            }
        } else {
            if (grp == 0) {
#pragma unroll
                for (int q = 0; q < 5; ++q) {
                    float* col = &Tq[q][mlo * CSTR + 8];
                    *(v4f*)(col)     = __builtin_shufflevector(acc[q], acc[q], 0, 1, 2, 3);
                    *(v4f*)(col + 4) = __builtin_shufflevector(acc[q], acc[q], 4, 5, 6, 7);
                }
            } else {
                const v2f z2 = {0.f, 0.f};
#pragma unroll
                for (int q = 0; q < 5; ++q) {
                    float* col = &Tq[q][mlo * CSTR + 24];
                    *(v2f*)(col)     = __builtin_shufflevector(acc[q], acc[q], 0, 1);
                    *(v2f*)(col + 2) = z2;      // zero pad k=26,27
                }
            }
        }
    }

    // ================= Vertical pass: out = Gv * T =================
    // out[m][n] = sum_k g[k-m] * T[k][n]; B loads unconditional (pad zeros).
    v8f vacc[5];
#pragma unroll
    for (int q = 0; q < 5; ++q) vacc[q] = zero8;

#pragma unroll
    for (int c = 0; c < 7; ++c) {
        const int kb = (c << 2) + koff;
#pragma unroll
        for (int q = 0; q < 5; ++q) {
            v2f bq = *(const v2f*)&Tq[q][mlo * CSTR + kb];
            vacc[q] = WMMA4(wco[c], bq, vacc[q]);
        }
    }

    // ================= SSIM map + reduction =================
    const float C1 = 0.01f * 0.01f;
    const float C2 = 0.03f * 0.03f;
    float lsum = 0.0f;
#pragma unroll
    for (int j = 0; j < 8; ++j) {
        float mu1  = vacc[0][j];
        float mu2  = vacc[1][j];
        float w11  = vacc[2][j];
        float w22  = vacc[3][j];
        float w12  = vacc[4][j];
        float mu1s = mu1 * mu1;
        float mu2s = mu2 * mu2;
        float mu12 = mu1 * mu2;
        float s1   = w11 - mu1s;
        float s2   = w22 - mu2s;
        float s12  = w12 - mu12;
        float num  = (2.0f * mu12 + C1) * (2.0f * s12 + C2);
        float den  = (mu1s + mu2s + C1) * (s1 + s2 + C2);
        lsum += num / den;
    }

    // wave32 reduction
#pragma unroll
    for (int off = 16; off > 0; off >>= 1)
        lsum += __shfl_xor(lsum, off, 32);

    if (l == 0)
        atomicAdd(acc_out, (double)lsum);
}

extern "C" void kernel_launch(void* const* d_in, const int* in_sizes, int n_in,
                              void* d_out, int out_size, void* d_ws, size_t ws_size,
                              hipStream_t stream) {
    const float* img1 = (const float*)d_in[0];
    const float* img2 = (const float*)d_in[1];
    float* out = (float*)d_out;
    double* acc = (double*)d_ws;

    ssim_init_kernel<<<1, 1, 0, stream>>>(acc);
    ssim_main_kernel<<<dim3(NBLOCKS), dim3(32), 0, stream>>>(img1, img2, acc);
    ssim_final_kernel<<<1, 1, 0, stream>>>(acc, out);
}